// BiLSTM_ATTN_ADVR_CRF_30236569764173
// MI455X (gfx1250) — compile-verified
//
#include <hip/hip_runtime.h>
#include <hip/hip_bf16.h>
#include <math.h>

// ---------------------------------------------------------------------------
// BiLSTM + general-attention + CRF NLL, fp32, MI455X (gfx1250, wave32, WMMA)
//
// Dims: T=1024, E=300, H=512, 4H=2048, D=2H=1024, V=100000, K=12
// GEMMs use V_WMMA_F32_16X16X4_F32 with a 32x64 register-blocked macro-tile
// per wave (8 accumulators): 8 WMMAs per 6 fragment loads -> load-issue
// pressure drops ~2.7x versus 1-tile-per-wave. Workload is L2-resident and
// dominated by the serial LSTM/CRF scans, so fp32 WMMA (exact reference
// precision) is the right choice over bf16/fp8.
// ---------------------------------------------------------------------------

#define Tn   1024
#define En   300
#define Hn   512
#define H4n  2048
#define Dn   1024
#define Kn   12
#define START_TAG 10
#define STOP_TAG  11

typedef __attribute__((ext_vector_type(2))) float v2f;
typedef __attribute__((ext_vector_type(8))) float v8f;

// ---------------------------------------------------------------------------
// 1) Embedding gather: x[t,:] = word_embeds[sentence[t],:]
// ---------------------------------------------------------------------------
__global__ void gather_embed(const int* __restrict__ sent,
                             const float* __restrict__ we,
                             float* __restrict__ x) {
  const int t = blockIdx.x;
  const long w = (long)sent[t];
  for (int e = threadIdx.x; e < En; e += blockDim.x)
    x[t * En + e] = we[w * En + e];
}

// ---------------------------------------------------------------------------
// 2) Transpose Whh [2048,512] -> WhhT [512,2048] (coalesced scan reads)
// ---------------------------------------------------------------------------
__global__ void transpose_whh(const float* __restrict__ W,
                              float* __restrict__ Wt) {
  const int g = blockIdx.x * 16 + threadIdx.x;  // 0..2047
  const int k = blockIdx.y * 16 + threadIdx.y;  // 0..511
  if (g < H4n && k < Hn) Wt[k * H4n + g] = W[g * Hn + k];
}

// ---------------------------------------------------------------------------
// 3) fp32 WMMA GEMM:  C[M,N] = act( A[M,K] * op(B) + bias )
//    transB=1: op(B)=B^T, B stored [N,K]; transB=0: op(B)=B, B stored [K,N].
//    Register blocking: each wave owns a 32x64 C macro-tile
//      (2 M-subtiles x 4 N-subtiles, 8 x v8f accumulators).
//    Per k-step (K+=4): 2 A-frag + 4 B-frag loads feed 8 WMMAs.
//    Requirements: K%4==0, M%32==0.  N masked/clamped (for N=12 head).
//    Fragment layouts per CDNA5 ISA 7.12.2 (wave32):
//      A 16x4 : lanes0-15 {M=lane, K=r},   lanes16-31 {M=lane-16, K=2+r}
//      B 4x16 : lanes0-15 {K=r, N=lane},   lanes16-31 {K=2+r, N=lane-16}
//      C 16x16: vgpr r -> {M=r(+8 for hi lanes), N=lane&15}
// ---------------------------------------------------------------------------
#define WPB 4   // waves per block (128 threads); each wave: 32 rows x 64 cols

__global__ void wmma_gemm(const float* __restrict__ A,
                          const float* __restrict__ B,
                          const float* __restrict__ bias,
                          float* __restrict__ C,
                          int M, int N, int Kd,
                          int lda, int ldb, int ldc,
                          int transB, int act) {
  const int lane = threadIdx.x & 31;
  const int wave = threadIdx.x >> 5;
  const int m0 = blockIdx.y << 5;                    // 32-row macro tile
  const int n0 = (blockIdx.x * WPB + wave) << 6;     // 64-col macro tile
  const int Npad = (N + 63) & ~63;
  if (n0 >= Npad) return;                 // wave-uniform: EXEC stays all-ones

  const int mrow  = lane & 15;
  const int khalf = (lane >> 4) << 1;     // 0 or 2

  const float* __restrict__ arow0 = A + (size_t)(m0 + mrow) * lda + khalf;
  const float* __restrict__ arow1 = arow0 + (size_t)16 * lda;

  v8f acc[2][4];
#pragma unroll
  for (int i = 0; i < 2; ++i)
#pragma unroll
    for (int j = 0; j < 4; ++j)
      acc[i][j] = (v8f){0.f, 0.f, 0.f, 0.f, 0.f, 0.f, 0.f, 0.f};

  if (transB) {
    const float* __restrict__ brow[4];
#pragma unroll
    for (int j = 0; j < 4; ++j) {
      int nn = n0 + 16 * j + mrow;
      if (nn > N - 1) nn = N - 1;         // clamp (only hit when N%64 != 0)
      brow[j] = B + (size_t)nn * ldb + khalf;
    }
#pragma unroll 2
    for (int k = 0; k < Kd; k += 4) {
      v2f a0, a1, b[4];
      a0.x = arow0[k]; a0.y = arow0[k + 1];
      a1.x = arow1[k]; a1.y = arow1[k + 1];
#pragma unroll
      for (int j = 0; j < 4; ++j) { b[j].x = brow[j][k]; b[j].y = brow[j][k + 1]; }
#pragma unroll
      for (int j = 0; j < 4; ++j) {
        acc[0][j] = __builtin_amdgcn_wmma_f32_16x16x4_f32(false, a0, false, b[j],
                                                          (short)0, acc[0][j], false, false);
        acc[1][j] = __builtin_amdgcn_wmma_f32_16x16x4_f32(false, a1, false, b[j],
                                                          (short)0, acc[1][j], false, false);
      }
    }
  } else {
    const int nnb = n0 + mrow;            // N%64==0 on this path
#pragma unroll 2
    for (int k = 0; k < Kd; k += 4) {
      v2f a0, a1, b[4];
      a0.x = arow0[k]; a0.y = arow0[k + 1];
      a1.x = arow1[k]; a1.y = arow1[k + 1];
      const float* __restrict__ br0 = B + (size_t)(k + khalf) * ldb + nnb;
      const float* __restrict__ br1 = br0 + ldb;
#pragma unroll
      for (int j = 0; j < 4; ++j) { b[j].x = br0[16 * j]; b[j].y = br1[16 * j]; }
#pragma unroll
      for (int j = 0; j < 4; ++j) {
        acc[0][j] = __builtin_amdgcn_wmma_f32_16x16x4_f32(false, a0, false, b[j],
                                                          (short)0, acc[0][j], false, false);
        acc[1][j] = __builtin_amdgcn_wmma_f32_16x16x4_f32(false, a1, false, b[j],
                                                          (short)0, acc[1][j], false, false);
      }
    }
  }

  const int rhalf = (lane >> 4) << 3;     // +8 for hi lanes
#pragma unroll
  for (int j = 0; j < 4; ++j) {
    const int col = n0 + 16 * j + (lane & 15);
    if (col < N) {
      const float bv = bias ? bias[col] : 0.f;
#pragma unroll
      for (int i = 0; i < 2; ++i) {
        const int rbase = m0 + 16 * i + rhalf;
#pragma unroll
        for (int r = 0; r < 8; ++r) {
          float v = acc[i][j][r] + bv;
          if (act) v = tanhf(v);
          C[(size_t)(rbase + r) * ldc + col] = v;
        }
      }
    }
  }
}

// ---------------------------------------------------------------------------
// 4) LSTM scan (sequential over T).  blockIdx.x = direction (0 fwd, 1 bwd).
//    1024 threads; thread tid owns gate preacts {tid, tid+1024}; threads
//    0..511 own cell state c[tid].  h lives in LDS; gates exchanged via LDS.
//    Output written to cat[:, dir*512 : dir*512+512] (row stride 2048).
// ---------------------------------------------------------------------------
__global__ void lstm_scan(const float* __restrict__ xg_f,
                          const float* __restrict__ xg_b,
                          const float* __restrict__ wtf,
                          const float* __restrict__ wtb,
                          const float* __restrict__ bhh_f,
                          const float* __restrict__ bhh_b,
                          float* __restrict__ cat) {
  const int dir = blockIdx.x;
  const float* __restrict__ xg = dir ? xg_b : xg_f;
  const float* __restrict__ wt = dir ? wtb : wtf;
  const float* __restrict__ bh = dir ? bhh_b : bhh_f;
  const int tid = threadIdx.x;  // 0..1023

  __shared__ float sh_h[Hn];
  __shared__ float sh_g[H4n];

  float c = 0.f;
  if (tid < Hn) sh_h[tid] = 0.f;
  const float b0 = bh[tid];
  const float b1 = bh[tid + 1024];
  __syncthreads();

  for (int step = 0; step < Tn; ++step) {
    const int t = dir ? (Tn - 1 - step) : step;
    const float* __restrict__ xr = xg + (size_t)t * H4n;
    float a0 = xr[tid] + b0;
    float a1 = xr[tid + 1024] + b1;
#pragma unroll 8
    for (int k = 0; k < Hn; ++k) {
      const float hk = sh_h[k];                   // LDS broadcast
      a0 = fmaf(hk, wt[(size_t)k * H4n + tid],        a0);
      a1 = fmaf(hk, wt[(size_t)k * H4n + tid + 1024], a1);
    }
    sh_g[tid] = a0;
    sh_g[tid + 1024] = a1;
    __syncthreads();
    if (tid < Hn) {
      const float ig = 1.f / (1.f + expf(-sh_g[tid]));
      const float fg = 1.f / (1.f + expf(-sh_g[Hn + tid]));
      const float gg = tanhf(sh_g[2 * Hn + tid]);
      const float og = 1.f / (1.f + expf(-sh_g[3 * Hn + tid]));
      c = fg * c + ig * gg;
      const float h = og * tanhf(c);
      sh_h[tid] = h;
      cat[(size_t)t * (2 * Dn) + dir * Hn + tid] = h;
    }
    __syncthreads();
  }
}

// ---------------------------------------------------------------------------
// 5) Row softmax over scores [T,T], in place.
// ---------------------------------------------------------------------------
__global__ void softmax_rows(float* __restrict__ S) {
  const int row = blockIdx.x;
  const int tid = threadIdx.x;  // 256
  __shared__ float red[256];
  float* __restrict__ r = S + (size_t)row * Tn;

  float m = -INFINITY;
  for (int j = tid; j < Tn; j += 256) m = fmaxf(m, r[j]);
  red[tid] = m;
  __syncthreads();
  for (int s = 128; s > 0; s >>= 1) {
    if (tid < s) red[tid] = fmaxf(red[tid], red[tid + s]);
    __syncthreads();
  }
  m = red[0];
  __syncthreads();

  float sum = 0.f;
  for (int j = tid; j < Tn; j += 256) {
    const float e = expf(r[j] - m);
    r[j] = e;
    sum += e;
  }
  red[tid] = sum;
  __syncthreads();
  for (int s = 128; s > 0; s >>= 1) {
    if (tid < s) red[tid] += red[tid + s];
    __syncthreads();
  }
  const float inv = 1.f / red[0];
  for (int j = tid; j < Tn; j += 256) r[j] *= inv;
}

// ---------------------------------------------------------------------------
// 6) CRF forward (logZ) + gold-path score; out[0] = logZ - gold.  One wave.
// ---------------------------------------------------------------------------
__global__ void crf_nll(const float* __restrict__ feats,
                        const float* __restrict__ trans,
                        const int* __restrict__ tags,
                        float* __restrict__ out) {
  const int lane = threadIdx.x;  // 32 threads
  __shared__ float alpha[16];
  __shared__ float tr[Kn * Kn];
  __shared__ float red[32];

  for (int i = lane; i < Kn * Kn; i += 32) tr[i] = trans[i];
  if (lane < Kn) alpha[lane] = (lane == START_TAG) ? 0.f : -10000.f;
  __syncthreads();

  for (int t = 0; t < Tn; ++t) {
    float na = 0.f;
    if (lane < Kn) {
      float m = -INFINITY;
#pragma unroll
      for (int f = 0; f < Kn; ++f) m = fmaxf(m, alpha[f] + tr[lane * Kn + f]);
      float s = 0.f;
#pragma unroll
      for (int f = 0; f < Kn; ++f) s += expf(alpha[f] + tr[lane * Kn + f] - m);
      na = m + logf(s) + feats[t * Kn + lane];
    }
    __syncthreads();
    if (lane < Kn) alpha[lane] = na;
    __syncthreads();
  }

  float logZ = 0.f;
  if (lane == 0) {
    float m = -INFINITY;
#pragma unroll
    for (int to = 0; to < Kn; ++to)
      m = fmaxf(m, alpha[to] + tr[STOP_TAG * Kn + to]);
    float s = 0.f;
#pragma unroll
    for (int to = 0; to < Kn; ++to)
      s += expf(alpha[to] + tr[STOP_TAG * Kn + to] - m);
    logZ = m + logf(s);
  }

  // gold = sum_t trans[pad_stop[t], pad_start[t]] + sum_t feats[t, tags[t]]
  float g = 0.f;
  for (int t = lane; t < Tn; t += 32) g += feats[t * Kn + tags[t]];
  for (int t = lane; t <= Tn; t += 32) {
    const int from = (t == 0)  ? START_TAG : tags[t - 1];
    const int to   = (t == Tn) ? STOP_TAG  : tags[t];
    g += tr[to * Kn + from];
  }
  red[lane] = g;
  __syncthreads();
  if (lane == 0) {
    float gold = 0.f;
    for (int i = 0; i < 32; ++i) gold += red[i];
    out[0] = logZ - gold;
  }
}

// ---------------------------------------------------------------------------
// Host-side orchestration
// ---------------------------------------------------------------------------
extern "C" void kernel_launch(void* const* d_in, const int* in_sizes, int n_in,
                              void* d_out, int out_size, void* d_ws, size_t ws_size,
                              hipStream_t stream) {
  (void)in_sizes; (void)n_in; (void)out_size; (void)ws_size;

  const int*   sentence = (const int*)  d_in[0];
  const int*   tags     = (const int*)  d_in[1];
  const float* word_emb = (const float*)d_in[2];
  const float* Wih_f    = (const float*)d_in[3];
  const float* Whh_f    = (const float*)d_in[4];
  const float* bih_f    = (const float*)d_in[5];
  const float* bhh_f    = (const float*)d_in[6];
  const float* Wih_b    = (const float*)d_in[7];
  const float* Whh_b    = (const float*)d_in[8];
  const float* bih_b    = (const float*)d_in[9];
  const float* bhh_b    = (const float*)d_in[10];
  const float* attn_W   = (const float*)d_in[11];
  const float* attn_b   = (const float*)d_in[12];
  const float* h2h1_W   = (const float*)d_in[13];
  const float* h2h1_b   = (const float*)d_in[14];
  const float* tag_W    = (const float*)d_in[15];
  const float* tag_b    = (const float*)d_in[16];
  const float* transit  = (const float*)d_in[17];
  float* out = (float*)d_out;

  // Workspace layout (floats)
  float* ws = (float*)d_ws;
  float* x      = ws;                       // [1024, 300]
  float* xg_f   = x      + Tn * En;         // [1024, 2048]
  float* xg_b   = xg_f   + Tn * H4n;        // [1024, 2048]
  float* wtf    = xg_b   + Tn * H4n;        // [512, 2048]
  float* wtb    = wtf    + Hn * H4n;        // [512, 2048]
  float* cat    = wtb    + Hn * H4n;        // [1024, 2048]  (enc | ctx)
  float* proj   = cat    + Tn * 2 * Dn;     // [1024, 1024]
  float* scores = proj   + Tn * Dn;         // [1024, 1024]
  float* h1     = scores + Tn * Tn;         // [1024, 1024]
  float* feats  = h1     + Tn * Dn;         // [1024, 12]

  // 1) x = word_embeds[sentence]
  gather_embed<<<Tn, 256, 0, stream>>>(sentence, word_emb, x);

  // 2) Transpose Whh for coalesced scan reads
  {
    dim3 b(16, 16), g(H4n / 16, Hn / 16);
    transpose_whh<<<g, b, 0, stream>>>(Whh_f, wtf);
    transpose_whh<<<g, b, 0, stream>>>(Whh_b, wtb);
  }

  // Per-wave macro tile: 32 rows x 64 cols; block covers 32 x (64*WPB).
  const int colsPerBlock = 64 * WPB;  // 256

  // 3) Input projections: xg = x @ Wih^T + bih   [1024,2048]
  {
    dim3 b(32 * WPB), g(H4n / colsPerBlock, Tn / 32);
    wmma_gemm<<<g, b, 0, stream>>>(x, Wih_f, bih_f, xg_f,
                                   Tn, H4n, En, En, En, H4n, 1, 0);
    wmma_gemm<<<g, b, 0, stream>>>(x, Wih_b, bih_b, xg_b,
                                   Tn, H4n, En, En, En, H4n, 1, 0);
  }

  // 4) Bidirectional LSTM scans -> cat[:, 0:1024] = enc
  lstm_scan<<<2, 1024, 0, stream>>>(xg_f, xg_b, wtf, wtb, bhh_f, bhh_b, cat);

  // 5) proj = enc @ attn_W^T + attn_b
  {
    dim3 b(32 * WPB), g(Dn / colsPerBlock, Tn / 32);
    wmma_gemm<<<g, b, 0, stream>>>(cat, attn_W, attn_b, proj,
                                   Tn, Dn, Dn, 2 * Dn, Dn, Dn, 1, 0);
  }

  // 6) scores = enc @ proj^T
  {
    dim3 b(32 * WPB), g(Tn / colsPerBlock, Tn / 32);
    wmma_gemm<<<g, b, 0, stream>>>(cat, proj, nullptr, scores,
                                   Tn, Tn, Dn, 2 * Dn, Dn, Tn, 1, 0);
  }

  // 7) softmax rows
  softmax_rows<<<Tn, 256, 0, stream>>>(scores);

  // 8) ctx = w @ enc  -> cat[:, 1024:2048]
  {
    dim3 b(32 * WPB), g(Dn / colsPerBlock, Tn / 32);
    wmma_gemm<<<g, b, 0, stream>>>(scores, cat, nullptr, cat + Dn,
                                   Tn, Dn, Tn, Tn, 2 * Dn, 2 * Dn, 0, 0);
  }

  // 9) h1 = tanh([enc|ctx] @ h2h1_W^T + b)
  {
    dim3 b(32 * WPB), g(Dn / colsPerBlock, Tn / 32);
    wmma_gemm<<<g, b, 0, stream>>>(cat, h2h1_W, h2h1_b, h1,
                                   Tn, Dn, 2 * Dn, 2 * Dn, 2 * Dn, Dn, 1, 1);
  }

  // 10) feats = h1 @ tag_W^T + tag_b   [1024, 12]
  {
    dim3 b(32 * WPB), g(1, Tn / 32);
    wmma_gemm<<<g, b, 0, stream>>>(h1, tag_W, tag_b, feats,
                                   Tn, Kn, Dn, Dn, Dn, Kn, 1, 0);
  }

  // 11) CRF: out[0] = logZ - gold
  crf_nll<<<1, 32, 0, stream>>>(feats, transit, tags, out);
}